// SubGraph_57037165691506
// MI455X (gfx1250) — compile-verified
//
#include <hip/hip_runtime.h>

typedef __attribute__((ext_vector_type(2))) float v2f;
typedef __attribute__((ext_vector_type(8))) float v8f;

#define FMID 128
#define NHEAD 4
#define CDIM 32
#define NEG_SLOPE 0.2f
#define AS_LOCAL __attribute__((address_space(3)))
#define USE_ASYNC_TILE 1

// ---------------- helpers ----------------
__device__ __forceinline__ unsigned f2ord(float f) {
    unsigned u = __float_as_uint(f);
    return (u & 0x80000000u) ? ~u : (u | 0x80000000u);
}
__device__ __forceinline__ float ord2f(unsigned o) {
    unsigned u = (o & 0x80000000u) ? (o & 0x7fffffffu) : ~o;
    return __uint_as_float(u);
}

// ---------------- GEMM: Y[N,128] = X[N,K] @ W[K,128] + bias ----------------
// One wave computes a 16x16 tile with V_WMMA_F32_16X16X4_F32.
// Block = 256 threads = 8 waves -> 16 rows x 128 cols per block.
// X tile staged in LDS via CDNA5 async global->LDS b128 copies.
template <int K>
__global__ __launch_bounds__(256) void gemm_bias_wmma(
    const float* __restrict__ X, const float* __restrict__ W,
    const float* __restrict__ bias, float* __restrict__ Y, int N)
{
    constexpr int PITCH = K + 4;          // 16B-aligned rows; pitch%64==4 -> no bank conflicts
    __shared__ float lx[16 * PITCH];
    const int row0 = blockIdx.x * 16;
    const int t = threadIdx.x;

    // stage 16 x K tile of X into LDS, 16 bytes per async transfer
    constexpr int NQUAD = 4 * K;          // number of float4s in the tile
#pragma unroll
    for (int q = t; q < NQUAD; q += 256) {
        const int r  = q / (K / 4);
        const int c4 = (q % (K / 4)) * 4;
        const float* g = X + (size_t)(row0 + r) * K + c4;
#if USE_ASYNC_TILE
        unsigned ldsAddr =
            (unsigned)(unsigned long long)(AS_LOCAL char*)&lx[r * PITCH + c4];
        asm volatile("global_load_async_to_lds_b128 %0, %1, off"
                     :: "v"(ldsAddr), "v"(g) : "memory");
#else
        const float4 v = *(const float4*)g;
        lx[r * PITCH + c4 + 0] = v.x;
        lx[r * PITCH + c4 + 1] = v.y;
        lx[r * PITCH + c4 + 2] = v.z;
        lx[r * PITCH + c4 + 3] = v.w;
#endif
    }
#if USE_ASYNC_TILE
    asm volatile("s_wait_asynccnt 0" ::: "memory");
#endif
    __syncthreads();

    const int lane = t & 31;
    const int half = lane >> 4;           // 0: K pair (k,k+1), 1: (k+2,k+3)
    const int lm   = lane & 15;
    const int col0 = (t >> 5) * 16;       // wave id * 16

    v8f acc = {0.f, 0.f, 0.f, 0.f, 0.f, 0.f, 0.f, 0.f};

#pragma unroll 8
    for (int k = 0; k < K; k += 4) {
        const int ka = k + 2 * half;
        v2f a, b;
        // A 16x4: lane lm holds row M=lm; VGPR0/1 = K=ka, ka+1
        a.x = lx[lm * PITCH + ka];
        a.y = lx[lm * PITCH + ka + 1];
        // B 4x16: lane lm holds col N=col0+lm; VGPR0/1 = K=ka, ka+1
        b.x = W[(size_t)ka * FMID + col0 + lm];
        b.y = W[(size_t)(ka + 1) * FMID + col0 + lm];
        acc = __builtin_amdgcn_wmma_f32_16x16x4_f32(
            false, a, false, b, (short)0, acc, false, false);
    }

    const float bv = bias[col0 + lm];
#pragma unroll
    for (int i = 0; i < 8; ++i) {
        const int row = row0 + i + 8 * half;   // C/D: VGPR i -> rows i, i+8
        Y[(size_t)row * FMID + col0 + lm] = acc[i] + bv;
    }
}

// ---------------- init per-layer accumulators ----------------
__global__ void init_buffers(unsigned* __restrict__ m_ord, float* __restrict__ z,
                             float* __restrict__ agg, int N)
{
    const int i = blockIdx.x * blockDim.x + threadIdx.x;
    if (i < N * NHEAD) { m_ord[i] = 0u; z[i] = 0.f; }  // 0 orders below every real
    if (i < N * FMID)  { agg[i] = 0.f; }
}

// ---------------- edge logits + segment max (wave per edge) ----------------
__global__ __launch_bounds__(256) void edge_logits(
    const float* __restrict__ xl, const float* __restrict__ xr,
    const float* __restrict__ att,
    const int* __restrict__ src, const int* __restrict__ dst,
    int E, int Etot, float* __restrict__ logits, unsigned* __restrict__ m_ord)
{
    const int e = blockIdx.x * 8 + (threadIdx.x >> 5);
    if (e >= Etot) return;
    const int lane = threadIdx.x & 31;
    int s, d;
    if (e < E) { s = src[e]; d = dst[e]; } else { s = e - E; d = s; }  // self-loop

    const float* pl = xl + (size_t)s * FMID;
    const float* pr = xr + (size_t)d * FMID;

    float acc[NHEAD];
#pragma unroll
    for (int h = 0; h < NHEAD; ++h) {
        const int f = h * CDIM + lane;          // lane covers c within each head
        float v = pl[f] + pr[f];
        v = (v > 0.f) ? v : NEG_SLOPE * v;      // LeakyReLU(0.2)
        acc[h] = v * att[f];                    // att[h][c]
    }
#pragma unroll
    for (int mask = 16; mask >= 1; mask >>= 1) {
#pragma unroll
        for (int h = 0; h < NHEAD; ++h)
            acc[h] += __shfl_xor(acc[h], mask, 32);
    }
    if (lane < NHEAD) {
        logits[(size_t)e * NHEAD + lane] = acc[lane];
        atomicMax(&m_ord[(size_t)d * NHEAD + lane], f2ord(acc[lane]));
    }
}

// ------- fused exp + segment-sum + weighted scatter-add (wave per edge) -----
// agg[d] += a * xl[s], z[d] += a; normalization deferred to finalize:
// sum(a*x)/ (sum(a)+eps) == sum((a/(sum(a)+eps)) * x)
__global__ __launch_bounds__(256) void edge_combine(
    const float* __restrict__ xl, const float* __restrict__ logits,
    const unsigned* __restrict__ m_ord,
    const int* __restrict__ src, const int* __restrict__ dst,
    int E, int Etot, float* __restrict__ z, float* __restrict__ agg)
{
    const int e = blockIdx.x * 8 + (threadIdx.x >> 5);
    if (e >= Etot) return;
    const int lane = threadIdx.x & 31;
    int s, d;
    if (e < E) { s = src[e]; d = dst[e]; } else { s = e - E; d = s; }

    float a[NHEAD];
#pragma unroll
    for (int h = 0; h < NHEAD; ++h) {
        a[h] = expf(logits[(size_t)e * NHEAD + h] -
                    ord2f(m_ord[(size_t)d * NHEAD + h]));
    }
    if (lane < NHEAD)
        atomicAdd(&z[(size_t)d * NHEAD + lane], a[lane]);

    const float* pl = xl + (size_t)s * FMID;
    float* pd = agg + (size_t)d * FMID;
#pragma unroll
    for (int h = 0; h < NHEAD; ++h) {
        const int f = h * CDIM + lane;
        atomicAdd(&pd[f], pl[f] * a[h]);        // global_atomic_add_f32
    }
}

// ---------------- fused normalize + bias + ReLU ----------------
__global__ void finalize_relu(const float* __restrict__ agg,
                              const float* __restrict__ z,
                              const float* __restrict__ bias,
                              float* __restrict__ out, int total)
{
    const int i = blockIdx.x * blockDim.x + threadIdx.x;
    if (i >= total) return;
    // i = node*128 + head*32 + c  ->  (node*4 + head) == i >> 5
    const float v = agg[i] / (z[i >> 5] + 1e-16f) + bias[i & (FMID - 1)];
    out[i] = (v > 0.f) ? v : 0.f;
}

// ---------------- orchestration ----------------
extern "C" void kernel_launch(void* const* d_in, const int* in_sizes, int n_in,
                              void* d_out, int out_size, void* d_ws, size_t ws_size,
                              hipStream_t stream)
{
    const float* x     = (const float*)d_in[0];
    const int*   ei    = (const int*)  d_in[1];   // [2, E] row-major
    const float* Wl1   = (const float*)d_in[2];
    const float* Wr1   = (const float*)d_in[3];
    const float* bl1   = (const float*)d_in[4];
    const float* br1   = (const float*)d_in[5];
    const float* att1  = (const float*)d_in[6];
    const float* bias1 = (const float*)d_in[7];
    const float* Wl2   = (const float*)d_in[8];
    const float* Wr2   = (const float*)d_in[9];
    const float* bl2   = (const float*)d_in[10];
    const float* br2   = (const float*)d_in[11];
    const float* att2  = (const float*)d_in[12];
    const float* bias2 = (const float*)d_in[13];

    const int N    = in_sizes[0] / 64;   // 100000
    const int E    = in_sizes[1] / 2;    // 1600000
    const int Etot = E + N;              // + self loops
    const int* src = ei;
    const int* dst = ei + E;

    const size_t nf = (size_t)N * FMID;
    float*    XL = (float*)d_ws;
    float*    XR = XL + nf;
    float*    AG = XR + nf;              // agg, then layer output in-place
    float*    LG = AG + nf;              // [Etot, 4] raw logits
    unsigned* MO = (unsigned*)(LG + (size_t)Etot * NHEAD);
    float*    Z  = (float*)(MO + (size_t)N * NHEAD);

    const dim3 blk(256);
    const int gGemm = N / 16;                       // N divisible by 16
    const int gWave = (Etot + 7) / 8;               // wave-per-edge
    const int gNode = (int)((nf + 255) / 256);

    // ---------------- layer 1 (K = 64) ----------------
    gemm_bias_wmma<64><<<gGemm, blk, 0, stream>>>(x, Wl1, bl1, XL, N);
    gemm_bias_wmma<64><<<gGemm, blk, 0, stream>>>(x, Wr1, br1, XR, N);
    init_buffers<<<gNode, blk, 0, stream>>>(MO, Z, AG, N);
    edge_logits<<<gWave, blk, 0, stream>>>(XL, XR, att1, src, dst, E, Etot, LG, MO);
    edge_combine<<<gWave, blk, 0, stream>>>(XL, LG, MO, src, dst, E, Etot, Z, AG);
    finalize_relu<<<gNode, blk, 0, stream>>>(AG, Z, bias1, AG, (int)nf);  // in-place -> H1

    // ---------------- layer 2 (K = 128) ----------------
    gemm_bias_wmma<128><<<gGemm, blk, 0, stream>>>(AG, Wl2, bl2, XL, N);
    gemm_bias_wmma<128><<<gGemm, blk, 0, stream>>>(AG, Wr2, br2, XR, N);
    init_buffers<<<gNode, blk, 0, stream>>>(MO, Z, AG, N);  // AG free after GEMMs
    edge_logits<<<gWave, blk, 0, stream>>>(XL, XR, att2, src, dst, E, Etot, LG, MO);
    edge_combine<<<gWave, blk, 0, stream>>>(XL, LG, MO, src, dst, E, Etot, Z, AG);
    finalize_relu<<<gNode, blk, 0, stream>>>(AG, Z, bias2, (float*)d_out, (int)nf);
}